// SimpleGNNModel_8830452760704
// MI455X (gfx1250) — compile-verified
//
#include <hip/hip_runtime.h>

#define HID 128
#define NN 50000
#define NE 800000

typedef __attribute__((ext_vector_type(2))) float v2f;
typedef __attribute__((ext_vector_type(8))) float v8f;

// ---------------- utility kernels ----------------

__global__ void zero_kernel(float* __restrict__ p, int n) {
  int i = blockIdx.x * blockDim.x + threadIdx.x;
  int stride = gridDim.x * blockDim.x;
  for (; i < n; i += stride) p[i] = 0.0f;
}

__global__ void degree_kernel(const int* __restrict__ src, const int* __restrict__ dst,
                              float* __restrict__ deg_out, float* __restrict__ deg_in,
                              int n_edges) {
  int e = blockIdx.x * blockDim.x + threadIdx.x;
  if (e < n_edges) {
    atomicAdd(deg_out + src[e], 1.0f);
    atomicAdd(deg_in + dst[e], 1.0f);
  }
}

__global__ void norm_kernel(float* __restrict__ deg, int n) {
  int i = blockIdx.x * blockDim.x + threadIdx.x;
  if (i < n) deg[i] = rsqrtf(fmaxf(deg[i], 1.0f));
}

// h[n, j] = node_feats[n, 0:4] @ W[0:4, j] + b[j]
__global__ void node_embed_kernel(const float* __restrict__ nf, const float* __restrict__ W,
                                  const float* __restrict__ b, float* __restrict__ h,
                                  int n_nodes) {
  int idx = blockIdx.x * blockDim.x + threadIdx.x;
  if (idx >= n_nodes * HID) return;
  int n = idx >> 7, j = idx & 127;
  const float* x = nf + n * 4;
  float acc = b[j];
  acc += x[0] * W[0 * HID + j];
  acc += x[1] * W[1 * HID + j];
  acc += x[2] * W[2 * HID + j];
  acc += x[3] * W[3 * HID + j];
  h[idx] = acc;
}

// agg[dst[e], f] += h[src[e], f] * norm_out[src[e]]
__global__ void scatter_kernel(const float* __restrict__ h, const float* __restrict__ norm_out,
                               const int* __restrict__ src, const int* __restrict__ dst,
                               float* __restrict__ agg) {
  int idx = blockIdx.x * blockDim.x + threadIdx.x;  // covers NE*128 exactly
  int e = idx >> 7, f = idx & 127;
  int s = src[e], d = dst[e];
  atomicAdd(agg + d * HID + f, h[s * HID + f] * norm_out[s]);
}

// ---------------- WMMA GEMM kernels ----------------
// Wave owns 16 rows x 128 cols. A (16x128 f32) resident in 64 VGPRs.
// f32 WMMA 16x16x4 A layout: lane l -> M = l&15; VGPR0/1 hold K = (l>>4)*2, +1
// f32 WMMA 16x16x4 B layout: lane l -> N = l&15; VGPR0/1 hold K = (l>>4)*2, +1
// C/D layout: VGPR r, lane l -> M = r + (l>>4)*8, N = l&15

// hout = relu((agg * norm_in[:,None]) @ W + b)   [n_nodes x 128]
__global__ void __launch_bounds__(256)
layer_gemm_kernel(const float* __restrict__ agg, const float* __restrict__ norm_in,
                  const float* __restrict__ W, const float* __restrict__ b,
                  float* __restrict__ hout, int n_nodes) {
  __shared__ float lds_wt[HID * HID];  // transposed: [n][k]
  __shared__ float lds_b[HID];
  int tid = threadIdx.x;
  for (int i = tid; i < HID * HID; i += 256) {
    int k = i >> 7, n = i & 127;
    lds_wt[n * HID + k] = W[i];
  }
  if (tid < HID) lds_b[tid] = b[tid];
  __syncthreads();

  int wave = tid >> 5, lane = tid & 31;
  int m = lane & 15, half = lane >> 4, kk = half * 2;
  int base = (blockIdx.x * 8 + wave) * 16;
  if (base >= n_nodes) return;

  int node = base + m;
  float nin = norm_in[node];
  const float* arow = agg + (size_t)node * HID;

  v2f A[32];
#pragma unroll
  for (int kb = 0; kb < 32; ++kb) {
    float2 t = *(const float2*)(arow + kb * 4 + kk);
    v2f a;
    a.x = t.x * nin;
    a.y = t.y * nin;
    A[kb] = a;
  }

  for (int t = 0; t < 8; ++t) {
    int nb = t * 16 + m;  // this lane's output column / B's N index
    v8f acc = {};
#pragma unroll
    for (int kb = 0; kb < 32; ++kb) {
      float2 bw = *(const float2*)(lds_wt + nb * HID + kb * 4 + kk);
      v2f bb;
      bb.x = bw.x;
      bb.y = bw.y;
      acc = __builtin_amdgcn_wmma_f32_16x16x4_f32(false, A[kb], false, bb,
                                                  (short)0, acc, false, false);
    }
    float bias = lds_b[nb];
#pragma unroll
    for (int r = 0; r < 8; ++r) {
      float v = acc[r] + bias;
      v = v > 0.0f ? v : 0.0f;
      hout[(size_t)(base + half * 8 + r) * HID + nb] = v;
    }
  }
}

// out[e] = relu((h[src[e]]+h[dst[e]]) @ W1 + b1) @ W2 + b2   (fused, OUT_DIM=1)
__global__ void __launch_bounds__(256)
edge_mlp_kernel(const float* __restrict__ h, const int* __restrict__ src,
                const int* __restrict__ dst, const float* __restrict__ W1,
                const float* __restrict__ b1, const float* __restrict__ W2,
                const float* __restrict__ b2, float* __restrict__ out, int n_edges) {
  __shared__ float lds_w1t[HID * HID];  // transposed: [n][k]
  __shared__ float lds_b1[HID];
  __shared__ float lds_w2[HID];
  int tid = threadIdx.x;
  for (int i = tid; i < HID * HID; i += 256) {
    int k = i >> 7, n = i & 127;
    lds_w1t[n * HID + k] = W1[i];
  }
  if (tid < HID) {
    lds_b1[tid] = b1[tid];
    lds_w2[tid] = W2[tid];
  }
  __syncthreads();
  float b2v = b2[0];

  int wave = tid >> 5, lane = tid & 31;
  int m = lane & 15, half = lane >> 4, kk = half * 2;
  int ebase = (blockIdx.x * 8 + wave) * 16;
  if (ebase >= n_edges) return;

  int e = ebase + m;
  const float* hs = h + (size_t)src[e] * HID;
  const float* hd = h + (size_t)dst[e] * HID;

  v2f A[32];
#pragma unroll
  for (int kb = 0; kb < 32; ++kb) {
    float2 a0 = *(const float2*)(hs + kb * 4 + kk);
    float2 a1 = *(const float2*)(hd + kb * 4 + kk);
    v2f a;
    a.x = a0.x + a1.x;
    a.y = a0.y + a1.y;
    A[kb] = a;
  }

  float pr[8];
#pragma unroll
  for (int r = 0; r < 8; ++r) pr[r] = 0.0f;

  for (int t = 0; t < 8; ++t) {
    int nb = t * 16 + m;
    v8f acc = {};
#pragma unroll
    for (int kb = 0; kb < 32; ++kb) {
      float2 bw = *(const float2*)(lds_w1t + nb * HID + kb * 4 + kk);
      v2f bb;
      bb.x = bw.x;
      bb.y = bw.y;
      acc = __builtin_amdgcn_wmma_f32_16x16x4_f32(false, A[kb], false, bb,
                                                  (short)0, acc, false, false);
    }
    float bias = lds_b1[nb], w2v = lds_w2[nb];
#pragma unroll
    for (int r = 0; r < 8; ++r) {
      float v = acc[r] + bias;
      v = v > 0.0f ? v : 0.0f;
      pr[r] += v * w2v;  // partial of the 128->1 dot product
    }
  }

  // reduce partial dots across the 16 lanes of each half (butterfly)
#pragma unroll
  for (int off = 1; off < 16; off <<= 1) {
#pragma unroll
    for (int r = 0; r < 8; ++r) pr[r] += __shfl_xor(pr[r], off, 32);
  }

  int rr = lane & 15;
  if (rr < 8) {
    float v = pr[0];
#pragma unroll
    for (int r = 1; r < 8; ++r) v = (rr == r) ? pr[r] : v;
    out[ebase + half * 8 + rr] = v + b2v;
  }
}

// ---------------- launcher ----------------

extern "C" void kernel_launch(void* const* d_in, const int* in_sizes, int n_in,
                              void* d_out, int out_size, void* d_ws, size_t ws_size,
                              hipStream_t stream) {
  (void)in_sizes; (void)n_in; (void)out_size; (void)ws_size;
  const float* node_feats = (const float*)d_in[1];
  const int*   src        = (const int*)d_in[2];
  const int*   dst        = (const int*)d_in[3];
  const float* W_nemb     = (const float*)d_in[6];
  const float* b_nemb     = (const float*)d_in[7];
  const float* gnn_W      = (const float*)d_in[8];
  const float* gnn_b      = (const float*)d_in[9];
  const float* out_W1     = (const float*)d_in[10];
  const float* out_b1     = (const float*)d_in[11];
  const float* out_W2     = (const float*)d_in[12];
  const float* out_b2     = (const float*)d_in[13];
  float* out = (float*)d_out;

  float* ws       = (float*)d_ws;
  float* norm_out = ws;                       // NN floats
  float* norm_in  = ws + NN;                  // NN floats
  float* h        = ws + 2 * NN;              // NN*128 floats
  float* agg      = h + (size_t)NN * HID;     // NN*128 floats

  // degrees -> norms (norm stored in place of degree)
  zero_kernel<<<256, 256, 0, stream>>>(norm_out, 2 * NN);
  degree_kernel<<<(NE + 255) / 256, 256, 0, stream>>>(src, dst, norm_out, norm_in, NE);
  norm_kernel<<<(2 * NN + 255) / 256, 256, 0, stream>>>(norm_out, 2 * NN);

  // node embedding
  node_embed_kernel<<<(NN * HID + 255) / 256, 256, 0, stream>>>(node_feats, W_nemb, b_nemb, h, NN);

  // two GraphConv layers: scatter-sum then WMMA GEMM (+bias, relu), ping-pong h<->agg
  for (int l = 0; l < 2; ++l) {
    zero_kernel<<<2048, 256, 0, stream>>>(agg, NN * HID);
    scatter_kernel<<<(NE * HID) / 256, 256, 0, stream>>>(h, norm_out, src, dst, agg);
    layer_gemm_kernel<<<(NN / 16 + 7) / 8, 256, 0, stream>>>(
        agg, norm_in, gnn_W + l * HID * HID, gnn_b + l * HID, h, NN);
  }

  // fused edge MLP: gather + GEMM(128x128) + relu + dot(128->1) + bias
  edge_mlp_kernel<<<NE / 16 / 8, 256, 0, stream>>>(h, src, dst, out_W1, out_b1,
                                                   out_W2, out_b2, out, NE);
}